// GraphAttentionNetwork_30657476558877
// MI455X (gfx1250) — compile-verified
//
#include <hip/hip_runtime.h>
#include <hip/hip_bf16.h>
#include <math.h>

#define NN 25000
#define NE 400000
#define DF 128
#define HID 64
#define HEADS 4
#define D1 (HID * HEADS)  // 256

typedef __attribute__((ext_vector_type(2))) float v2f;
typedef __attribute__((ext_vector_type(8))) float v8f;

__device__ __forceinline__ float leaky02(float x) { return x > 0.f ? x : 0.2f * x; }
__device__ __forceinline__ float elu1(float x) { return x > 0.f ? x : (__expf(x) - 1.f); }

// Monotonic-bits float atomic max (correct for mixed signs).
__device__ __forceinline__ void atomic_max_f32(float* addr, float v) {
  if (v >= 0.f) {
    atomicMax((int*)addr, __float_as_int(v));
  } else {
    atomicMin((unsigned int*)addr, __float_as_uint(v));
  }
}

// ---------------------------------------------------------------------------
// H[n, 64] = X[n, kdim] @ W[kdim, 64] + bias, via V_WMMA_F32_16X16X4_F32.
// One wave per 16x16 output tile; 4 waves/block cover the 64 output columns.
// Fragment layouts per CDNA5 ISA 7.12.2.
// ---------------------------------------------------------------------------
__global__ __launch_bounds__(128)
void gat_gemm_wmma(const float* __restrict__ X, const float* __restrict__ W,
                   const float* __restrict__ bias, float* __restrict__ H,
                   int n_rows, int kdim) {
  const int wave = threadIdx.x >> 5;
  const int lane = threadIdx.x & 31;
  const int m0 = blockIdx.x * 16;
  const int n0 = wave * 16;
  const int lrow = lane & 15;
  const int khalf = (lane >> 4) * 2;  // 0 or 2

  int arow = m0 + lrow;
  if (arow >= n_rows) arow = n_rows - 1;  // clamp loads; stores are guarded
  const int bcol = n0 + lrow;

  const float bv = bias[bcol];
  v8f acc;
#pragma unroll
  for (int i = 0; i < 8; ++i) acc[i] = bv;

  for (int k = 0; k < kdim; k += 4) {
    v2f a, b;
    a.x = X[(size_t)arow * kdim + k + khalf + 0];
    a.y = X[(size_t)arow * kdim + k + khalf + 1];
    b.x = W[(size_t)(k + khalf + 0) * HID + bcol];
    b.y = W[(size_t)(k + khalf + 1) * HID + bcol];
    acc = __builtin_amdgcn_wmma_f32_16x16x4_f32(
        /*neg_a=*/false, a, /*neg_b=*/false, b,
        /*c_mod=*/(short)0, acc, /*reuse_a=*/false, /*reuse_b=*/false);
  }

  const int roff = (lane >> 4) * 8;
#pragma unroll
  for (int v = 0; v < 8; ++v) {
    const int row = m0 + v + roff;
    if (row < n_rows) H[(size_t)row * HID + bcol] = acc[v];
  }
}

// ---------------------------------------------------------------------------
// Segment buffers: [NN][HID][HEADS] fp32 (segmax, S0 = sum of exp, S1 = sum of exp*sent)
// ---------------------------------------------------------------------------
__global__ __launch_bounds__(256)
void gat_init_seg(float* __restrict__ segmax, float* __restrict__ s0,
                  float* __restrict__ s1, int n) {
  const int i = blockIdx.x * blockDim.x + threadIdx.x;
  if (i < n) {
    segmax[i] = -3.0e38f;
    s0[i] = 0.f;
    s1[i] = 0.f;
  }
}

// Pass 1: per (edge, hid-pair) compute head logits, atomic-max into segmax[recv].
// Thread t handles edge e = t>>5, hidden channels {2*(t&31), 2*(t&31)+1}.
__global__ __launch_bounds__(256)
void gat_edge_max(const float* __restrict__ H, const int* __restrict__ snd,
                  const int* __restrict__ rcv, const float* __restrict__ Wl,
                  const float* __restrict__ bl, float* __restrict__ segmax) {
  const long long t = (long long)blockIdx.x * blockDim.x + threadIdx.x;
  if (t >= (long long)NE * (HID / 2)) return;
  const int e = (int)(t >> 5);
  const int hid = (int)(t & 31) << 1;
  const int s = snd[e], r = rcv[e];
  const float2 sh = *(const float2*)(H + (size_t)s * HID + hid);
  const float2 rh = *(const float2*)(H + (size_t)r * HID + hid);
  float* out = segmax + ((size_t)r * HID + hid) * HEADS;
#pragma unroll
  for (int h = 0; h < HEADS; ++h) {
    const float w0 = Wl[h], w1 = Wl[HEADS + h], b = bl[h];
    atomic_max_f32(out + h,         leaky02(fmaf(sh.x, w0, fmaf(rh.x, w1, b))));
    atomic_max_f32(out + HEADS + h, leaky02(fmaf(sh.y, w0, fmaf(rh.y, w1, b))));
  }
}

// Pass 2 (fused expsum + weighted aggregate, normalization deferred to node side):
//   ex = exp(logit - segmax);  S0 += ex;  S1 += ex * sent
__global__ __launch_bounds__(256)
void gat_edge_accum(const float* __restrict__ H, const int* __restrict__ snd,
                    const int* __restrict__ rcv, const float* __restrict__ Wl,
                    const float* __restrict__ bl,
                    const float* __restrict__ segmax,
                    float* __restrict__ s0, float* __restrict__ s1) {
  const long long t = (long long)blockIdx.x * blockDim.x + threadIdx.x;
  if (t >= (long long)NE * (HID / 2)) return;
  const int e = (int)(t >> 5);
  const int hid = (int)(t & 31) << 1;
  const int s = snd[e], r = rcv[e];
  const float2 sh = *(const float2*)(H + (size_t)s * HID + hid);
  const float2 rh = *(const float2*)(H + (size_t)r * HID + hid);
  const size_t idx = ((size_t)r * HID + hid) * HEADS;
#pragma unroll
  for (int h = 0; h < HEADS; ++h) {
    const float w0 = Wl[h], w1 = Wl[HEADS + h], b = bl[h];
    const float lg0 = leaky02(fmaf(sh.x, w0, fmaf(rh.x, w1, b)));
    const float lg1 = leaky02(fmaf(sh.y, w0, fmaf(rh.y, w1, b)));
    const float ex0 = __expf(lg0 - segmax[idx + h]);
    const float ex1 = __expf(lg1 - segmax[idx + HEADS + h]);
    atomicAdd(s0 + idx + h, ex0);
    atomicAdd(s0 + idx + HEADS + h, ex1);
    atomicAdd(s1 + idx + h, ex0 * sh.x);
    atomicAdd(s1 + idx + HEADS + h, ex1 * sh.y);
  }
}

// Layers 0/1: out[i] = elu(S1[i]/S0[i]); [NN][HID][HEADS] flat == reshape(N,256).
__global__ __launch_bounds__(256)
void gat_act_elu(const float* __restrict__ s0, const float* __restrict__ s1,
                 float* __restrict__ out, int n) {
  const int i = blockIdx.x * blockDim.x + threadIdx.x;
  if (i < n) {
    const float d = s0[i];
    const float a = d > 0.f ? s1[i] / d : 0.f;  // in-degree 0 -> 0 (matches reference)
    out[i] = elu1(a);
  }
}

// Layer 2: mean over heads of S1/S0, then ELU -> d_out [NN][HID].
__global__ __launch_bounds__(256)
void gat_act_mean_elu(const float* __restrict__ s0, const float* __restrict__ s1,
                      float* __restrict__ out) {
  const int i = blockIdx.x * blockDim.x + threadIdx.x;
  if (i >= NN * HID) return;
  const size_t base = (size_t)i * HEADS;
  float acc = 0.f;
#pragma unroll
  for (int h = 0; h < HEADS; ++h) {
    const float d = s0[base + h];
    acc += (d > 0.f ? s1[base + h] / d : 0.f);
  }
  out[i] = elu1(acc * 0.25f);
}

extern "C" void kernel_launch(void* const* d_in, const int* in_sizes, int n_in,
                              void* d_out, int out_size, void* d_ws, size_t ws_size,
                              hipStream_t stream) {
  (void)in_sizes; (void)n_in; (void)out_size; (void)ws_size;

  const float* nodes = (const float*)d_in[0];
  const int* snd = (const int*)d_in[1];
  const int* rcv = (const int*)d_in[2];
  const float* Wq[3] = {(const float*)d_in[3], (const float*)d_in[7], (const float*)d_in[11]};
  const float* bq[3] = {(const float*)d_in[4], (const float*)d_in[8], (const float*)d_in[12]};
  const float* Wl[3] = {(const float*)d_in[5], (const float*)d_in[9], (const float*)d_in[13]};
  const float* bl[3] = {(const float*)d_in[6], (const float*)d_in[10], (const float*)d_in[14]};

  // Workspace carve-out (fp32): H(1.6M) + X1(6.4M) + X2(6.4M) + segmax/S0/S1(6.4M each)
  float* ws = (float*)d_ws;
  float* H = ws;
  float* X1 = H + (size_t)NN * HID;
  float* X2 = X1 + (size_t)NN * D1;
  float* segmax = X2 + (size_t)NN * D1;
  float* s0 = segmax + (size_t)NN * D1;
  float* s1 = s0 + (size_t)NN * D1;

  const int segN = NN * D1;  // 6.4M elems
  const int initBlocks = (segN + 255) / 256;
  const long long eT = (long long)NE * (HID / 2);  // 12.8M threads
  const int edgeBlocks = (int)((eT + 255) / 256);
  const dim3 gemmGrid((NN + 15) / 16);
  const dim3 gemmBlock(128);

  const float* Xin = nodes;
  int kdim = DF;
  for (int layer = 0; layer < 3; ++layer) {
    gat_gemm_wmma<<<gemmGrid, gemmBlock, 0, stream>>>(Xin, Wq[layer], bq[layer], H, NN, kdim);
    gat_init_seg<<<initBlocks, 256, 0, stream>>>(segmax, s0, s1, segN);
    gat_edge_max<<<edgeBlocks, 256, 0, stream>>>(H, snd, rcv, Wl[layer], bl[layer], segmax);
    gat_edge_accum<<<edgeBlocks, 256, 0, stream>>>(H, snd, rcv, Wl[layer], bl[layer], segmax, s0, s1);
    if (layer < 2) {
      float* Xn = (layer == 0) ? X1 : X2;
      gat_act_elu<<<initBlocks, 256, 0, stream>>>(s0, s1, Xn, segN);
      Xin = Xn;
      kdim = D1;
    } else {
      gat_act_mean_elu<<<(NN * HID + 255) / 256, 256, 0, stream>>>(s0, s1, (float*)d_out);
    }
  }
}